// GCNConv_67619965108640
// MI455X (gfx1250) — compile-verified
//
#include <hip/hip_runtime.h>
#include <hip/hip_bf16.h>

typedef __attribute__((ext_vector_type(2))) float v2f;
typedef __attribute__((ext_vector_type(8))) float v8f;

#define D_DIM 64   // D_IN == D_OUT == 64

// -----------------------------------------------------------------------------
// Kernel 1: out[n][d] = bias[d]   (pre-initialize accumulator target)
// -----------------------------------------------------------------------------
__global__ __launch_bounds__(256)
void gcn_bias_init(const float* __restrict__ bias, float* __restrict__ out,
                   int total4) {
  int i = blockIdx.x * blockDim.x + threadIdx.x;   // one float4 per thread
  if (i >= total4) return;
  const float4* b4 = (const float4*)bias;          // 64 floats = 16 float4
  ((float4*)out)[i] = b4[i & 15];
}

// -----------------------------------------------------------------------------
// Kernel 2: support = X @ W via V_WMMA_F32_16X16X4_F32 (exact fp32)
// One wave -> 16 rows x 64 cols (4 col-tiles), K=64 in steps of 4.
// A 16x4 layout (ISA 7.12.2): lane<16 holds K=0,1 ; lane>=16 holds K=2,3.
// C/D layout: VGPR r -> M = r + 8*(lane>=16), N = lane&15.
// -----------------------------------------------------------------------------
__global__ __launch_bounds__(256)
void gcn_gemm_wmma(const float* __restrict__ X, const float* __restrict__ W,
                   float* __restrict__ S, int N) {
  const int wave    = threadIdx.x >> 5;                 // 0..7
  const int lane    = threadIdx.x & 31;
  const int rowTile = blockIdx.x * 8 + wave;            // 16 rows per tile
  const int m0      = rowTile * 16;
  if (m0 >= N) return;

  const int lhalf = lane >> 4;        // 0 | 1
  const int l16   = lane & 15;
  const int arow  = m0 + l16;
  const int koff  = lhalf * 2;        // A/B k sub-offset for this half-wave

  v8f acc0 = {}, acc1 = {}, acc2 = {}, acc3 = {};

  const float* __restrict__ arowp = X + (size_t)arow * D_DIM;

  #pragma unroll
  for (int k = 0; k < D_DIM; k += 4) {
    // A fragment: contiguous float2 per lane
    v2f a;
    a.x = arowp[k + koff + 0];
    a.y = arowp[k + koff + 1];

    // B fragments for the 4 column tiles: W[k+koff][n], W[k+koff+1][n]
    const float* __restrict__ brow = W + (size_t)(k + koff) * D_DIM + l16;
    #pragma unroll
    for (int ct = 0; ct < 4; ++ct) {
      v2f b;
      b.x = brow[ct * 16];
      b.y = brow[ct * 16 + D_DIM];
      v8f* accp = (ct == 0) ? &acc0 : (ct == 1) ? &acc1 : (ct == 2) ? &acc2 : &acc3;
      *accp = __builtin_amdgcn_wmma_f32_16x16x4_f32(
          /*neg_a=*/false, a, /*neg_b=*/false, b,
          /*c_mod=*/(short)0, *accp, /*reuse_a=*/false, /*reuse_b=*/false);
    }
  }

  // Store D tiles: row = m0 + r + 8*lhalf, col = ct*16 + l16
  float* __restrict__ outp = S + (size_t)(m0 + 8 * lhalf) * D_DIM + l16;
  #pragma unroll
  for (int r = 0; r < 8; ++r) {
    outp[(size_t)r * D_DIM + 0]  = acc0[r];
    outp[(size_t)r * D_DIM + 16] = acc1[r];
    outp[(size_t)r * D_DIM + 32] = acc2[r];
    outp[(size_t)r * D_DIM + 48] = acc3[r];
  }
}

// -----------------------------------------------------------------------------
// Kernel 3: COO SpMM scatter: out[dst] += support[src] * w   (f32 L2 atomics)
// blockDim = (16,16): 16 edges/block, each thread = 4 contiguous columns.
// -----------------------------------------------------------------------------
__global__ __launch_bounds__(256)
void gcn_edge_scatter(const float* __restrict__ S,
                      const int* __restrict__ edge_index,   // [2][E]: row0=dst
                      const float* __restrict__ ew,
                      float* __restrict__ out, int E) {
  const int e = blockIdx.x * 16 + threadIdx.y;
  if (e >= E) return;
  const int   dst = edge_index[e];
  const int   src = edge_index[E + e];
  const float w   = ew[e];

  const int col = threadIdx.x * 4;
  const float4 v = *(const float4*)(S + (size_t)src * D_DIM + col);
  float* __restrict__ o = out + (size_t)dst * D_DIM + col;
  atomicAdd(o + 0, v.x * w);
  atomicAdd(o + 1, v.y * w);
  atomicAdd(o + 2, v.z * w);
  atomicAdd(o + 3, v.w * w);
}

// -----------------------------------------------------------------------------
// Launch: inputs = { x[N*64], edge_index[2*E], edge_weight[E], weight[64*64],
//                    bias[64] };  out = [N*64] f32;  d_ws holds support.
// -----------------------------------------------------------------------------
extern "C" void kernel_launch(void* const* d_in, const int* in_sizes, int n_in,
                              void* d_out, int out_size, void* d_ws, size_t ws_size,
                              hipStream_t stream) {
  const float* x      = (const float*)d_in[0];
  const int*   eidx   = (const int*)d_in[1];
  const float* ew     = (const float*)d_in[2];
  const float* weight = (const float*)d_in[3];
  const float* bias   = (const float*)d_in[4];
  float*       out    = (float*)d_out;
  float*       support = (float*)d_ws;          // N*64 floats scratch

  const int N = in_sizes[0] / D_DIM;
  const int E = in_sizes[2];

  // 1) out = bias (broadcast)
  {
    int total4 = (N * D_DIM) / 4;
    int blocks = (total4 + 255) / 256;
    gcn_bias_init<<<blocks, 256, 0, stream>>>(bias, out, total4);
  }
  // 2) support = X @ W  (WMMA fp32)
  {
    int rowTiles = N / 16;                       // N=100000 -> 6250
    int blocks   = (rowTiles + 7) / 8;
    gcn_gemm_wmma<<<blocks, 256, 0, stream>>>(x, weight, support, N);
  }
  // 3) out += A @ support  (edge scatter, f32 atomics into L2-resident out)
  {
    int blocks = (E + 15) / 16;
    dim3 blk(16, 16);
    gcn_edge_scatter<<<blocks, blk, 0, stream>>>(support, eidx, ew, out, E);
  }
}